// GrangerCausality_57964878627119
// MI455X (gfx1250) — compile-verified
//
#include <hip/hip_runtime.h>
#include <hip/hip_bf16.h>

typedef __attribute__((ext_vector_type(2))) float v2f;
typedef __attribute__((ext_vector_type(8))) float v8f;

#define NN 2048
#define DIN 2048
#define LATENT 2048
#define PROJ 256
#define HH 1024
#define LAGS 12

// ---------------------------------------------------------------------------
// Generic GEMM: C[M,Nc] = act(A[M,K] @ W[K,Nc] + bias), f32 WMMA 16x16x4.
// Block = 256 threads = 8 waves; wave computes a 16x64 strip (4 accumulators).
// Grid: (Nc/64, M/128). All dims used here are multiples of (128, 64, 4).
// ---------------------------------------------------------------------------
__global__ __launch_bounds__(256) void gemm_wmma_f32(
    const float* __restrict__ A, const float* __restrict__ W,
    const float* __restrict__ bias, float* __restrict__ C,
    int M, int K, int Nc, int doRelu)
{
    const int lane = threadIdx.x & 31;
    const int wave = threadIdx.x >> 5;
    const int r    = lane & 15;   // row-in-tile for A, col-in-tile for B
    const int half = lane >> 4;   // K-half selector (0 or 1)
    const int rowBase = blockIdx.y * 128 + wave * 16;
    const int colBase = blockIdx.x * 64;

    v8f acc[4] = {};

    const float* __restrict__ arow = A + (size_t)(rowBase + r) * K + 2 * half;

    for (int k = 0; k < K; k += 4) {
        v2f a;
        a.x = arow[k];
        a.y = arow[k + 1];

        const float* __restrict__ w0 = W + (size_t)(k + 2 * half) * Nc + colBase + r;
        const float* __restrict__ w1 = w0 + Nc;

        v2f b0, b1, b2, b3;
        b0.x = w0[0];  b0.y = w1[0];
        b1.x = w0[16]; b1.y = w1[16];
        b2.x = w0[32]; b2.y = w1[32];
        b3.x = w0[48]; b3.y = w1[48];

        acc[0] = __builtin_amdgcn_wmma_f32_16x16x4_f32(false, a, false, b0, (short)0, acc[0], false, false);
        acc[1] = __builtin_amdgcn_wmma_f32_16x16x4_f32(false, a, false, b1, (short)0, acc[1], false, false);
        acc[2] = __builtin_amdgcn_wmma_f32_16x16x4_f32(false, a, false, b2, (short)0, acc[2], false, false);
        acc[3] = __builtin_amdgcn_wmma_f32_16x16x4_f32(false, a, false, b3, (short)0, acc[3], false, false);
    }

    // C/D layout: VGPR v -> M = v + 8*half, N = lane%16 (per 16x16 f32 table)
    #pragma unroll
    for (int cg = 0; cg < 4; ++cg) {
        const int col = colBase + cg * 16 + r;
        const float bv = bias ? bias[col] : 0.0f;
        #pragma unroll
        for (int v = 0; v < 8; ++v) {
            const int row = rowBase + v + 8 * half;
            float val = acc[cg][v] + bv;
            if (doRelu) val = fmaxf(val, 0.0f);
            C[(size_t)row * Nc + col] = val;
        }
    }
}

// ---------------------------------------------------------------------------
// wl[n,i] = sum_l (LAGS-l)/78 * lagged[n,l,i]
// ---------------------------------------------------------------------------
__global__ __launch_bounds__(256) void k_weighted_lag(
    const float* __restrict__ lagged, float* __restrict__ wl)
{
    const size_t idx = (size_t)blockIdx.x * 256 + threadIdx.x;
    const int n = (int)(idx >> 11);
    const int i = (int)(idx & 2047);
    const float* p = lagged + (size_t)n * LAGS * DIN + i;
    float s = 0.0f;
    #pragma unroll
    for (int l = 0; l < LAGS; ++l)
        s += ((float)(LAGS - l) * (1.0f / 78.0f)) * p[(size_t)l * DIN];
    wl[idx] = s;
}

// ---------------------------------------------------------------------------
// z / mean / log_var from q_params (qp[n, 0:2048]=mean, qp[n, 2048:4096]=lv)
// ---------------------------------------------------------------------------
__global__ __launch_bounds__(256) void k_reparam(
    const float* __restrict__ qp, const float* __restrict__ eps,
    float* __restrict__ z_out, float* __restrict__ mean_out,
    float* __restrict__ lv_out)
{
    const size_t idx = (size_t)blockIdx.x * 256 + threadIdx.x;
    const int n = (int)(idx >> 11);
    const int j = (int)(idx & 2047);
    const float m  = qp[(size_t)n * (2 * LATENT) + j];
    const float lv = qp[(size_t)n * (2 * LATENT) + LATENT + j];
    z_out[idx]    = m + expf(0.5f * lv) * eps[idx];
    mean_out[idx] = m;
    lv_out[idx]   = lv;
}

// ---------------------------------------------------------------------------
// Column sum over rows: dst[j] = sum_r src[r*cols + j]
// ---------------------------------------------------------------------------
__global__ __launch_bounds__(256) void k_colsum(
    const float* __restrict__ src, float* __restrict__ dst, int rows, int cols)
{
    const int j = blockIdx.x * 256 + threadIdx.x;
    if (j >= cols) return;
    float s = 0.0f;
    for (int r2 = 0; r2 < rows; ++r2) s += src[(size_t)r2 * cols + j];
    dst[j] = s;
}

__global__ __launch_bounds__(256) void k_zero(int* __restrict__ p, int n)
{
    const int i = blockIdx.x * 256 + threadIdx.x;
    if (i < n) p[i] = 0;
}

// ---------------------------------------------------------------------------
// CSR build over E real edges + NN self-loops (edge id >= E => self loop)
// ---------------------------------------------------------------------------
__global__ __launch_bounds__(256) void k_hist(
    const int* __restrict__ ei, int* __restrict__ counts, int E, int Etot)
{
    const int i = blockIdx.x * 256 + threadIdx.x;
    if (i >= Etot) return;
    const int dst = (i < E) ? ei[E + i] : (i - E);
    atomicAdd(&counts[dst], 1);
}

__global__ __launch_bounds__(256) void k_scan2048(
    const int* __restrict__ counts, int* __restrict__ offsets)
{
    __shared__ int carry[257];
    __shared__ int part[256];
    const int t = threadIdx.x;
    int loc[8]; int s = 0;
    #pragma unroll
    for (int k = 0; k < 8; ++k) { loc[k] = counts[t * 8 + k]; s += loc[k]; }
    part[t] = s;
    __syncthreads();
    if (t == 0) {
        int run = 0;
        for (int i = 0; i < 256; ++i) { carry[i] = run; run += part[i]; }
        carry[256] = run;
    }
    __syncthreads();
    int run = carry[t];
    #pragma unroll
    for (int k = 0; k < 8; ++k) { offsets[t * 8 + k] = run; run += loc[k]; }
    if (t == 255) offsets[2048] = carry[256];
}

__global__ __launch_bounds__(256) void k_scatter(
    const int* __restrict__ ei, const int* __restrict__ offsets,
    int* __restrict__ ctr, int* __restrict__ csr, int E, int Etot)
{
    const int i = blockIdx.x * 256 + threadIdx.x;
    if (i >= Etot) return;
    const int dst = (i < E) ? ei[E + i] : (i - E);
    const int pos = offsets[dst] + atomicAdd(&ctr[dst], 1);
    csr[pos] = i;
}

// ---------------------------------------------------------------------------
// e[edge,h] = sum_d leaky(xl[src,h,d] + xr[dst,h,d], 0.2) * att[h,d]
// one wave per edge, 8 waves per block
// ---------------------------------------------------------------------------
__global__ __launch_bounds__(256) void k_edge_logits(
    const int* __restrict__ ei, const float* __restrict__ xl,
    const float* __restrict__ xr, const float* __restrict__ att,
    float* __restrict__ ebuf, int E, int Etot)
{
    const int lane = threadIdx.x & 31;
    const int wave = threadIdx.x >> 5;
    const int edge = blockIdx.x * 8 + wave;
    if (edge >= Etot) return;
    const int src = (edge < E) ? ei[edge] : (edge - E);
    const int dst = (edge < E) ? ei[E + edge] : (edge - E);
    const float* __restrict__ ps = xl + (size_t)src * HH;
    const float* __restrict__ pd = xr + (size_t)dst * HH;
    #pragma unroll
    for (int h = 0; h < 8; ++h) {
        float s = 0.0f;
        #pragma unroll
        for (int d = 0; d < 128; d += 32) {
            const int dd = h * 128 + d + lane;
            const float v = ps[dd] + pd[dd];
            const float lr = (v > 0.0f) ? v : 0.2f * v;
            s += lr * att[h * 128 + d + lane];
        }
        #pragma unroll
        for (int off = 16; off > 0; off >>= 1) s += __shfl_xor(s, off, 32);
        if (lane == 0) ebuf[(size_t)edge * 8 + h] = s;
    }
}

// ---------------------------------------------------------------------------
// Per-dst GAT aggregation: softmax over incoming edges per head,
// out[dst, h*128+d] = relu( (sum_e w_e * xl[src_e, h*128+d]) / sum_e w_e + bias )
// One block per dst node.
// ---------------------------------------------------------------------------
__global__ __launch_bounds__(256) void k_gat_agg(
    const int* __restrict__ csr, const int* __restrict__ offsets,
    const int* __restrict__ ei, const float* __restrict__ ebuf,
    const float* __restrict__ xl, const float* __restrict__ bias,
    float* __restrict__ outp, int E)
{
    __shared__ float red[256];
    __shared__ float mh[8];
    __shared__ float sden[8];
    __shared__ int   edsh[32];
    __shared__ int   srcsh[32];
    __shared__ float wsh[256];

    const int dst = blockIdx.x;
    const int t   = threadIdx.x;
    const int start = offsets[dst];
    const int nE    = offsets[dst + 1] - start;

    // phase 1: per-head max (order-independent)
    for (int h = 0; h < 8; ++h) {
        float lm = -3.0e38f;
        for (int p = t; p < nE; p += 256)
            lm = fmaxf(lm, ebuf[(size_t)csr[start + p] * 8 + h]);
        red[t] = lm;
        __syncthreads();
        for (int s = 128; s > 0; s >>= 1) {
            if (t < s) red[t] = fmaxf(red[t], red[t + s]);
            __syncthreads();
        }
        if (t == 0) mh[h] = red[0];
        __syncthreads();
    }

    if (t < 8) sden[t] = 0.0f;
    __syncthreads();

    float acc[4] = {0.0f, 0.0f, 0.0f, 0.0f};

    for (int base = 0; base < nE; base += 32) {
        const int cnt = min(32, nE - base);
        if (t < cnt) {
            const int ed = csr[start + base + t];
            edsh[t]  = ed;
            srcsh[t] = (ed < E) ? ei[ed] : (ed - E);
        }
        __syncthreads();
        if (t < cnt * 8) {
            const int j = t >> 3, h = t & 7;
            wsh[t] = expf(ebuf[(size_t)edsh[j] * 8 + h] - mh[h]);
        }
        __syncthreads();
        if (t < 8) {
            float s = 0.0f;
            for (int j = 0; j < cnt; ++j) s += wsh[j * 8 + t];
            sden[t] += s;
        }
        for (int j = 0; j < cnt; ++j) {
            const float* __restrict__ xs = xl + (size_t)srcsh[j] * HH;
            #pragma unroll
            for (int k = 0; k < 4; ++k) {
                const int d = t + k * 256;
                acc[k] += wsh[j * 8 + (d >> 7)] * xs[d];
            }
        }
        __syncthreads();
    }

    #pragma unroll
    for (int k = 0; k < 4; ++k) {
        const int d = t + k * 256;
        const float v = acc[k] / sden[d >> 7] + bias[d];
        outp[(size_t)dst * HH + d] = fmaxf(v, 0.0f);
    }
}

// ---------------------------------------------------------------------------
// causal[i,j] = A_dir[i,j]*xrs[j] + A_ind[i,j]*zs[j]; adj = one_hot(argmax row)
// ---------------------------------------------------------------------------
__global__ __launch_bounds__(256) void k_final(
    const float* __restrict__ Ad, const float* __restrict__ Ai,
    const float* __restrict__ xrs, const float* __restrict__ zs,
    const float* __restrict__ gum, float* __restrict__ causal,
    float* __restrict__ adj)
{
    __shared__ float rv[256];
    __shared__ int   ri[256];
    const int i = blockIdx.x;
    const int t = threadIdx.x;
    float bv = -3.0e38f; int bi = NN;
    for (int j = t; j < NN; j += 256) {
        const size_t p = (size_t)i * NN + j;
        const float c = Ad[p] * xrs[j] + Ai[p] * zs[j];
        causal[p] = c;
        const float lg = c + gum[p];   // /tau is monotone; argmax unchanged
        if (lg > bv) { bv = lg; bi = j; }
    }
    rv[t] = bv; ri[t] = bi;
    __syncthreads();
    for (int s = 128; s > 0; s >>= 1) {
        if (t < s) {
            if (rv[t + s] > rv[t] || (rv[t + s] == rv[t] && ri[t + s] < ri[t])) {
                rv[t] = rv[t + s]; ri[t] = ri[t + s];
            }
        }
        __syncthreads();
    }
    const int best = ri[0];
    for (int j = t; j < NN; j += 256)
        adj[(size_t)i * NN + j] = (j == best) ? 1.0f : 0.0f;
}

// ---------------------------------------------------------------------------
extern "C" void kernel_launch(void* const* d_in, const int* in_sizes, int n_in,
                              void* d_out, int out_size, void* d_ws, size_t ws_size,
                              hipStream_t stream) {
    // Live inputs (setup_inputs dict order):
    const int*   edge_index = (const int*)  d_in[4];
    const float* lagged     = (const float*)d_in[5];
    const float* eps        = (const float*)d_in[6];
    const float* gumbel     = (const float*)d_in[7];
    const float* enc_Wv     = (const float*)d_in[22];
    const float* enc_bv     = (const float*)d_in[23];
    const float* enc_fc_W   = (const float*)d_in[24];
    const float* enc_fc_b   = (const float*)d_in[25];
    const float* dec_Wv     = (const float*)d_in[30];
    const float* dec_bv     = (const float*)d_in[31];
    const float* gat_d_Wl   = (const float*)d_in[32];
    const float* gat_d_bl   = (const float*)d_in[33];
    const float* gat_d_Wr   = (const float*)d_in[34];
    const float* gat_d_br   = (const float*)d_in[35];
    const float* gat_d_att  = (const float*)d_in[36];
    const float* gat_d_bias = (const float*)d_in[37];
    const float* dec_fc_W   = (const float*)d_in[38];
    const float* dec_fc_b   = (const float*)d_in[39];
    const float* A_direct   = (const float*)d_in[40];
    const float* A_indirect = (const float*)d_in[41];

    const int E    = in_sizes[4] / 2;
    const int Etot = E + NN;

    // Workspace layout
    float* wsf  = (float*)d_ws;
    float* wl   = wsf;                           // NN*DIN
    float* venc = wl   + (size_t)NN * DIN;       // NN*PROJ
    float* qp   = venc + (size_t)NN * PROJ;      // NN*2*LATENT
    float* xdec = qp   + (size_t)NN * 2 * LATENT;// NN*PROJ
    float* xld  = xdec + (size_t)NN * PROJ;      // NN*HH
    float* xrd  = xld  + (size_t)NN * HH;        // NN*HH
    float* gdec = xrd  + (size_t)NN * HH;        // NN*HH
    float* ebuf = gdec + (size_t)NN * HH;        // Etot*8
    int* counts  = (int*)(ebuf + (size_t)Etot * 8);
    int* ctr     = counts + NN;
    float* zsum  = (float*)(ctr + NN);
    float* xrsum = zsum + NN;
    int* offsets = (int*)(xrsum + NN);           // NN+1
    int* csr     = offsets + NN + 1;             // Etot

    // Output layout: z, mean, log_var, x_rec, causal_effect, adj (each NN*2048)
    float* out = (float*)d_out;
    const size_t NM = (size_t)NN * 2048;
    float* z_out    = out;
    float* mean_out = out + NM;
    float* lv_out   = out + 2 * NM;
    float* xrec_out = out + 3 * NM;
    float* causal   = out + 4 * NM;
    float* adj      = out + 5 * NM;

    const int eb = (Etot + 255) / 256;

    // 0) zero CSR counters (counts, ctr contiguous = 2*NN ints)
    k_zero<<<(2 * NN + 255) / 256, 256, 0, stream>>>(counts, 2 * NN);

    // 1) weighted lag collapse (the only HBM-heavy pass: 201 MB in, 16.8 MB out)
    k_weighted_lag<<<(NN * DIN) / 256, 256, 0, stream>>>(lagged, wl);

    // 2) encoder value path: venc = relu(wl @ enc_Wv + bv)    [2048,2048]x[2048,256]
    gemm_wmma_f32<<<dim3(PROJ / 64, NN / 128), 256, 0, stream>>>(
        wl, enc_Wv, enc_bv, venc, NN, DIN, PROJ, 1);

    // 3) qp = venc @ enc_fc_W + b                             [2048,256]x[256,4096]
    gemm_wmma_f32<<<dim3((2 * LATENT) / 64, NN / 128), 256, 0, stream>>>(
        venc, enc_fc_W, enc_fc_b, qp, NN, PROJ, 2 * LATENT, 0);

    // 4) z / mean / log_var straight into d_out
    k_reparam<<<(NN * LATENT) / 256, 256, 0, stream>>>(qp, eps, z_out, mean_out, lv_out);
    k_colsum<<<(LATENT + 255) / 256, 256, 0, stream>>>(z_out, zsum, NN, LATENT);

    // 5) decoder value path: xdec = relu(wl @ dec_Wv + bv)
    gemm_wmma_f32<<<dim3(PROJ / 64, NN / 128), 256, 0, stream>>>(
        wl, dec_Wv, dec_bv, xdec, NN, DIN, PROJ, 1);

    // 6) decoder GATv2 source/target transforms
    gemm_wmma_f32<<<dim3(HH / 64, NN / 128), 256, 0, stream>>>(
        xdec, gat_d_Wl, gat_d_bl, xld, NN, PROJ, HH, 0);
    gemm_wmma_f32<<<dim3(HH / 64, NN / 128), 256, 0, stream>>>(
        xdec, gat_d_Wr, gat_d_br, xrd, NN, PROJ, HH, 0);

    // 7) CSR over incoming edges (+self loops), edge logits, aggregation
    k_hist<<<eb, 256, 0, stream>>>(edge_index, counts, E, Etot);
    k_scan2048<<<1, 256, 0, stream>>>(counts, offsets);
    k_scatter<<<eb, 256, 0, stream>>>(edge_index, offsets, ctr, csr, E, Etot);
    k_edge_logits<<<(Etot + 7) / 8, 256, 0, stream>>>(
        edge_index, xld, xrd, gat_d_att, ebuf, E, Etot);
    k_gat_agg<<<NN, 256, 0, stream>>>(
        csr, offsets, edge_index, ebuf, xld, gat_d_bias, gdec, E);

    // 8) x_rec = gdec @ dec_fc_W + b  -> d_out               [2048,1024]x[1024,2048]
    gemm_wmma_f32<<<dim3(DIN / 64, NN / 128), 256, 0, stream>>>(
        gdec, dec_fc_W, dec_fc_b, xrec_out, NN, HH, DIN, 0);
    k_colsum<<<(DIN + 255) / 256, 256, 0, stream>>>(xrec_out, xrsum, NN, DIN);

    // 9) causal effect + hard gumbel-softmax one-hot
    k_final<<<NN, 256, 0, stream>>>(A_direct, A_indirect, xrsum, zsum, gumbel,
                                    causal, adj);
}